// AuxIVA_T_ISS_9680856285491
// MI455X (gfx1250) — compile-verified
//
#include <hip/hip_runtime.h>
#include <math.h>

// ---------------- problem constants (from reference) ----------------
#define Bc 2
#define Cc 4
#define Fc 257
#define Nc 512
#define Tc 3           // N_TAPS
#define EPSf 1e-5f

static constexpr int BCFN = Bc * Cc * Fc * Nc;  // 1,052,672
static constexpr int BCN  = Bc * Cc * Nc;       // 4,096

typedef float v2f __attribute__((ext_vector_type(2)));
typedef float v8f __attribute__((ext_vector_type(8)));

// branchless guarded load: returns p[n-dly] for n>=dly else 0, without branches
// (index clamped so the load address is always valid -> v_max_i32 + load + v_cndmask)
__device__ __forceinline__ float ld_guard(const float* __restrict__ p, int n, int dly)
{
    int i = n - dly;
    i = (i < 0) ? 0 : i;
    float v = p[i];
    return (n >= dly) ? v : 0.f;
}

// ---------------- prep: Xc = X (working copy) ----------------
__global__ void k_prep(const float* __restrict__ Xr, const float* __restrict__ Xi,
                       float* __restrict__ Xcr, float* __restrict__ Xci)
{
    int i = blockIdx.x * blockDim.x + threadIdx.x;
    if (i < BCFN) { Xcr[i] = Xr[i]; Xci[i] = Xi[i]; }
}

// ---------------- per-iteration stats: S[b,c,n] = sum_f |Xc|^2 ----------------
__global__ __launch_bounds__(256) void k_stat(const float* __restrict__ Xcr,
                                              const float* __restrict__ Xci,
                                              float* __restrict__ S)
{
    int i = blockIdx.x * 256 + threadIdx.x;
    if (i >= BCN) return;
    int n = i % Nc, bc = i / Nc;
    const float* pr = Xcr + (size_t)bc * Fc * Nc + n;
    const float* pi = Xci + (size_t)bc * Fc * Nc + n;
    float acc = 0.f;
    for (int f = 0; f < Fc; ++f) {
        float r = pr[(size_t)f * Nc];
        float q = pi[(size_t)f * Nc];
        acc += r * r + q * q;
    }
    S[i] = acc;
}

// ---------------- weights + gain: one block per (b,c) ----------------
// g   = clip(mean |Xc|^2 over (F,N), eps)
// w   = g / clip(2*sqrt(S), eps)      (laplace weight * g)
// isg = 1/sqrt(g)                     (Xc scaling applied at load in k_iss)
__global__ __launch_bounds__(256) void k_weights(const float* __restrict__ S,
                                                 float* __restrict__ wgt,
                                                 float* __restrict__ isg)
{
    __shared__ float red[256];
    const int bc = blockIdx.x, tid = threadIdx.x;
    float a = S[bc * Nc + tid] + S[bc * Nc + tid + 256];
    red[tid] = a;
    __syncthreads();
    for (int s = 128; s > 0; s >>= 1) {
        if (tid < s) red[tid] += red[tid + s];
        __syncthreads();
    }
    const float g = fmaxf(red[0] / (float)(Fc * Nc), EPSf);
    if (tid == 0) isg[bc] = 1.0f / sqrtf(g);
    for (int n = tid; n < Nc; n += 256)
        wgt[bc * Nc + n] = g / fmaxf(2.0f * sqrtf(S[bc * Nc + n]), EPSf);
}

// ---------------- ISS updates: one 256-thread block per (b,f) ----------------
__global__ __launch_bounds__(256) void k_iss(const float* __restrict__ Xr,
                                             const float* __restrict__ Xi,
                                             float* __restrict__ Xcr_g,
                                             float* __restrict__ Xci_g,
                                             const float* __restrict__ wgt,
                                             const float* __restrict__ isg)
{
    __shared__ float xcr[Cc][Nc], xci[Cc][Nc];   // 16 KB  (current Xc for this f)
    __shared__ float wl[Cc][Nc];                 //  8 KB  (weights)
    __shared__ float xstr[Nc], xsti[Nc];         //  4 KB  (delayed tap row)
    __shared__ float red[12][256];               // 12 KB  (block reductions)

    const int bf = blockIdx.x;
    const int b = bf / Fc, f = bf % Fc, tid = threadIdx.x;

    // load Xc (applying this iteration's 1/sqrt(g) scale) and weights
    for (int c = 0; c < Cc; ++c) {
        const float s = isg[b * Cc + c];
        const size_t base = (size_t)((b * Cc + c) * Fc + f) * Nc;
        for (int n = tid; n < Nc; n += 256) {
            xcr[c][n] = Xcr_g[base + n] * s;
            xci[c][n] = Xci_g[base + n] * s;
            wl[c][n]  = wgt[(b * Cc + c) * Nc + n];
        }
    }
    __syncthreads();

    const float norm = 1.0f / (float)Nc;

    // ---- source rounds ----
    for (int src = 0; src < Cc; ++src) {
        float pr_[4] = {0, 0, 0, 0}, pi_[4] = {0, 0, 0, 0}, pd_[4] = {0, 0, 0, 0};
        for (int n = tid; n < Nc; n += 256) {
            float xsr = xcr[src][n], xsi = xci[src][n];
            float m2 = xsr * xsr + xsi * xsi;
#pragma unroll
            for (int c = 0; c < Cc; ++c) {
                float wv = wl[c][n];
                float xr = xcr[c][n], xq = xci[c][n];
                pr_[c] += wv * (xr * xsr + xq * xsi);   // Re(Xc * conj(Xs))
                pi_[c] += wv * (xq * xsr - xr * xsi);   // Im(Xc * conj(Xs))
                pd_[c] += wv * m2;
            }
        }
#pragma unroll
        for (int c = 0; c < Cc; ++c) {
            red[c][tid] = pr_[c]; red[4 + c][tid] = pi_[c]; red[8 + c][tid] = pd_[c];
        }
        __syncthreads();
        for (int s = 128; s > 0; s >>= 1) {
            if (tid < s) {
#pragma unroll
                for (int q = 0; q < 12; ++q) red[q][tid] += red[q][tid + s];
            }
            __syncthreads();
        }
        float vr[4], vi[4];
#pragma unroll
        for (int c = 0; c < Cc; ++c) {
            float dn = fmaxf(red[8 + c][0] * norm, EPSf);
            vr[c] = (red[c][0] * norm) / dn;
            vi[c] = (red[4 + c][0] * norm) / dn;
        }
        {
            float ds = fmaxf(red[8 + src][0] * norm, EPSf);
            vr[src] = 1.0f - 1.0f / sqrtf(ds);
            vi[src] = 0.0f;
        }
        for (int n = tid; n < Nc; n += 256) {
            float xsr = xcr[src][n], xsi = xci[src][n];
#pragma unroll
            for (int c = 0; c < Cc; ++c) {
                xcr[c][n] -= vr[c] * xsr - vi[c] * xsi;
                xci[c][n] -= vr[c] * xsi + vi[c] * xsr;
            }
        }
        __syncthreads();
    }

    // ---- dereverberation tap rounds (X_bar built on the fly from original X) ----
    for (int src = 0; src < Cc; ++src) {
        const float* pr0 = Xr + (size_t)((b * Cc + src) * Fc + f) * Nc;
        const float* pi0 = Xi + (size_t)((b * Cc + src) * Fc + f) * Nc;
        for (int tap = 0; tap < Tc; ++tap) {
            const int dly = (Tc + 1) - tap;   // delay = n_taps + n_delay - tap
            for (int n = tid; n < Nc; n += 256) {
                xstr[n] = ld_guard(pr0, n, dly);
                xsti[n] = ld_guard(pi0, n, dly);
            }
            float pr_[4] = {0, 0, 0, 0}, pi_[4] = {0, 0, 0, 0}, pd_[4] = {0, 0, 0, 0};
            for (int n = tid; n < Nc; n += 256) {
                float xsr = xstr[n], xsi = xsti[n];
                float m2 = xsr * xsr + xsi * xsi;
#pragma unroll
                for (int c = 0; c < Cc; ++c) {
                    float wv = wl[c][n];
                    float xr = xcr[c][n], xq = xci[c][n];
                    pr_[c] += wv * (xr * xsr + xq * xsi);
                    pi_[c] += wv * (xq * xsr - xr * xsi);
                    pd_[c] += wv * m2;
                }
            }
#pragma unroll
            for (int c = 0; c < Cc; ++c) {
                red[c][tid] = pr_[c]; red[4 + c][tid] = pi_[c]; red[8 + c][tid] = pd_[c];
            }
            __syncthreads();
            for (int s = 128; s > 0; s >>= 1) {
                if (tid < s) {
#pragma unroll
                    for (int q = 0; q < 12; ++q) red[q][tid] += red[q][tid + s];
                }
                __syncthreads();
            }
            float vr[4], vi[4];
#pragma unroll
            for (int c = 0; c < Cc; ++c) {
                float dn = fmaxf(red[8 + c][0], EPSf);   // no 1/N factor here (ref)
                vr[c] = red[c][0] / dn;
                vi[c] = red[4 + c][0] / dn;
            }
            for (int n = tid; n < Nc; n += 256) {
                float xsr = xstr[n], xsi = xsti[n];
#pragma unroll
                for (int c = 0; c < Cc; ++c) {
                    xcr[c][n] -= vr[c] * xsr - vi[c] * xsi;
                    xci[c][n] -= vr[c] * xsi + vi[c] * xsr;
                }
            }
            __syncthreads();
        }
    }

    // store back
    for (int c = 0; c < Cc; ++c) {
        const size_t base = (size_t)((b * Cc + c) * Fc + f) * Nc;
        for (int n = tid; n < Nc; n += 256) {
            Xcr_g[base + n] = xcr[c][n];
            Xci_g[base + n] = xci[c][n];
        }
    }
}

// ---------------- projection back: one wave32 per (b,f) ----------------
// covmat (16x16 complex Gram, K=512) via V_WMMA_F32_16X16X4_F32:
//   Re = Zr Zr^T + Zi Zi^T ;  Im = Zr Zi^T - Zi Zr^T   (A frag == B frag for a Gram)
// Only the first K-step can touch n < dly (dly <= 4): it is peeled with
// branchless clamped loads; the steady-state loop is unconditional.
__global__ __launch_bounds__(32) void k_proj(const float* __restrict__ Xr,
                                             const float* __restrict__ Xi,
                                             const float* __restrict__ Ycr,
                                             const float* __restrict__ Yci,
                                             float* __restrict__ out)
{
    __shared__ float MR[16][18], MI[16][18], FR[16], FI[16];
    const int bf = blockIdx.x;
    const int b = bf / Fc, f = bf % Fc;
    const int lane = threadIdx.x;
    const int m = lane & 15;            // Z row owned by this lane
    const int koff = (lane >> 4) << 1;  // K pair: lanes 0-15 -> k0,k0+1 ; 16-31 -> k0+2,k0+3

    const float *pr, *pi;
    int dly;
    if (m < 4) {  // rows 0..3: separated signal Y
        pr = Ycr + (size_t)((b * Cc + m) * Fc + f) * Nc;
        pi = Yci + (size_t)((b * Cc + m) * Fc + f) * Nc;
        dly = 0;
    } else {      // rows 4..15: X_bar row (d, u), delay = 4-u, from original X
        int mm = m - 4, d = mm / Tc, u = mm % Tc;
        dly = (Tc + 1) - u;
        pr = Xr + (size_t)((b * Cc + d) * Fc + f) * Nc;
        pi = Xi + (size_t)((b * Cc + d) * Fc + f) * Nc;
    }
    // pointers pre-shifted by the delay: element n of row m is prA[n] (valid for n >= dly)
    const float* prA = pr - dly;
    const float* piA = pi - dly;

    v8f accRe = {0.f, 0.f, 0.f, 0.f, 0.f, 0.f, 0.f, 0.f};
    v8f accIm = {0.f, 0.f, 0.f, 0.f, 0.f, 0.f, 0.f, 0.f};

    // ---- peeled first K-step (k0 = 0): n in {0..3}, guard against n < dly ----
    {
        int n0 = koff, n1 = koff + 1;
        v2f ar, ai, an;
        ar.x = ld_guard(pr, n0, dly);
        ar.y = ld_guard(pr, n1, dly);
        ai.x = ld_guard(pi, n0, dly);
        ai.y = ld_guard(pi, n1, dly);
        an.x = -ai.x; an.y = -ai.y;
        accRe = __builtin_amdgcn_wmma_f32_16x16x4_f32(false, ar, false, ar, (short)0, accRe, false, false);
        accRe = __builtin_amdgcn_wmma_f32_16x16x4_f32(false, ai, false, ai, (short)0, accRe, false, false);
        accIm = __builtin_amdgcn_wmma_f32_16x16x4_f32(false, ar, false, ai, (short)0, accIm, false, false);
        accIm = __builtin_amdgcn_wmma_f32_16x16x4_f32(false, an, false, ar, (short)0, accIm, false, false);
    }
    // ---- steady state: k0 >= 4 -> n0 >= 4 >= dly, unconditional loads ----
    for (int k0 = 4; k0 < Nc; k0 += 4) {
        int n0 = k0 + koff, n1 = n0 + 1;
        v2f ar, ai, an;
        ar.x = prA[n0]; ar.y = prA[n1];
        ai.x = piA[n0]; ai.y = piA[n1];
        an.x = -ai.x;   an.y = -ai.y;
        accRe = __builtin_amdgcn_wmma_f32_16x16x4_f32(false, ar, false, ar, (short)0, accRe, false, false);
        accRe = __builtin_amdgcn_wmma_f32_16x16x4_f32(false, ai, false, ai, (short)0, accRe, false, false);
        accIm = __builtin_amdgcn_wmma_f32_16x16x4_f32(false, ar, false, ai, (short)0, accIm, false, false);
        accIm = __builtin_amdgcn_wmma_f32_16x16x4_f32(false, an, false, ar, (short)0, accIm, false, false);
    }

    // scatter C/D fragments: lanes 0-15 -> rows v, col=lane ; lanes 16-31 -> rows v+8
    {
        const int col = lane & 15;
        const int rowB = (lane >> 4) * 8;
#pragma unroll
        for (int v = 0; v < 8; ++v) {
            MR[rowB + v][col] = accRe[v];
            MI[rowB + v][col] = accIm[v];
        }
    }

    // xcov column: xcov[c] = sum_n conj(Z[c,n]) * X[b,0,f,n]  (two lanes per row)
    {
        const int half = lane >> 4;
        const float* xmr = Xr + (size_t)((b * Cc + 0) * Fc + f) * Nc;
        const float* xmi = Xi + (size_t)((b * Cc + 0) * Fc + f) * Nc;
        float sr = 0.f, si = 0.f;
        // guarded region: n < 4
        for (int n = half; n < 4; n += 2) {
            float zr = ld_guard(pr, n, dly);
            float zi = ld_guard(pi, n, dly);
            float mr = xmr[n], mi = xmi[n];
            sr += zr * mr + zi * mi;
            si += zr * mi - zi * mr;
        }
        // unguarded region: n >= 4 >= dly
        for (int n = 4 + half; n < Nc; n += 2) {
            float zr = prA[n], zi = piA[n];
            float mr = xmr[n], mi = xmi[n];
            sr += zr * mr + zi * mi;
            si += zr * mi - zi * mr;
        }
        sr += __shfl_xor(sr, 16, 32);
        si += __shfl_xor(si, 16, 32);
        if (half == 0) { MR[m][16] = sr; MI[m][16] = si; }
    }
    __syncthreads();

    // Gauss-Jordan on the 16x17 augmented complex system (covmat is Hermitian PD)
    for (int k = 0; k < 16; ++k) {
        float prr = MR[k][k], pii = MI[k][k];
        float dd = fmaxf(prr * prr + pii * pii, 1e-30f);
        float ivr = prr / dd, ivi = -pii / dd;
        if (lane < 17) {
            float nr = MR[k][lane], ni = MI[k][lane];
            MR[k][lane] = nr * ivr - ni * ivi;
            MI[k][lane] = nr * ivi + ni * ivr;
        }
        __syncthreads();
        if (lane < 16) {
            FR[lane] = (lane == k) ? 0.f : MR[lane][k];
            FI[lane] = (lane == k) ? 0.f : MI[lane][k];
        }
        __syncthreads();
        for (int e = lane; e < 16 * 17; e += 32) {
            int i = e / 17, j = e % 17;
            if (i != k) {
                float fr = FR[i], fi = FI[i];
                float kr = MR[k][j], ki = MI[k][j];
                MR[i][j] -= fr * kr - fi * ki;
                MI[i][j] -= fr * ki + fi * kr;
            }
        }
        __syncthreads();
    }

    // Y_out = Y * a  (a = first 4 solution entries), write re/im planes
    float arc[4], aic[4];
#pragma unroll
    for (int c = 0; c < 4; ++c) { arc[c] = MR[c][16]; aic[c] = MI[c][16]; }
    for (int n = lane; n < Nc; n += 32) {
#pragma unroll
        for (int c = 0; c < 4; ++c) {
            size_t ix = (size_t)((b * Cc + c) * Fc + f) * Nc + n;
            float yr = Ycr[ix], yi = Yci[ix];
            out[ix]        = yr * arc[c] - yi * aic[c];
            out[BCFN + ix] = yr * aic[c] + yi * arc[c];
        }
    }
}

// ---------------- launch ----------------
extern "C" void kernel_launch(void* const* d_in, const int* in_sizes, int n_in,
                              void* d_out, int out_size, void* d_ws, size_t ws_size,
                              hipStream_t stream)
{
    (void)in_sizes; (void)n_in; (void)out_size; (void)ws_size;
    const float* Xr = (const float*)d_in[0];
    const float* Xi = (const float*)d_in[1];
    float* out = (float*)d_out;

    float* ws  = (float*)d_ws;
    float* Xcr = ws;                 // BCFN
    float* Xci = Xcr + BCFN;         // BCFN
    float* S   = Xci + BCFN;         // BCN
    float* wgt = S + BCN;            // BCN
    float* isg = wgt + BCN;          // Bc*Cc
    // total workspace: (2*BCFN + 2*BCN + 8) floats  ~= 8.1 MB

    k_prep<<<(BCFN + 255) / 256, 256, 0, stream>>>(Xr, Xi, Xcr, Xci);
    for (int it = 0; it < 4; ++it) {
        k_stat<<<(BCN + 255) / 256, 256, 0, stream>>>(Xcr, Xci, S);
        k_weights<<<Bc * Cc, 256, 0, stream>>>(S, wgt, isg);
        k_iss<<<Bc * Fc, 256, 0, stream>>>(Xr, Xi, Xcr, Xci, wgt, isg);
    }
    k_proj<<<Bc * Fc, 32, 0, stream>>>(Xr, Xi, Xcr, Xci, out);
}